// DySPN_Module_40338332844080
// MI455X (gfx1250) — compile-verified
//
#include <hip/hip_runtime.h>
#include <hip/hip_bf16.h>
#include <math.h>

// Problem constants (from reference)
#define Bn    2
#define Hn    240
#define Wn    1216
#define NPIX  (Bn * Hn * Wn)     // 583680
#define CHn   27                 // ITER*NUM
#define KDIM  243                // 27 in-channels * 9 taps
#define KPAD  244                // next multiple of 4
#define KSTEPS 61                // KPAD/4
#define MOUT  81                 // 3*CH output channels
#define MTILES 6                 // 96 padded channels / 16

// conv tiling
#define TW    64                 // pixels per workgroup (one row segment)
#define GS    66                 // LDS guide row stride = TW + 2 halo
#define NBX   (Wn / TW)          // 19 tiles per row (exact)

typedef __attribute__((ext_vector_type(2))) float v2f;
typedef __attribute__((ext_vector_type(8))) float v8f;

// ---------------------------------------------------------------------------
// Repack conv_w into per-lane WMMA A-fragment order (runs once, tiny):
//   apack[t*96 + m] = { A(m, 2t), A(m, 2t+1) },  t = 2*kstep + kh, t in [0,122)
//   A(m,k) = conv_w[m, k/9, (k%9)/3, (k%9)%3] for k<243 && m<81, else 0.
// 122*96 = 11712 float2 (93.7 KB), lives in d_ws after conv_out.
// ---------------------------------------------------------------------------
__global__ __launch_bounds__(256) void repack_weights_kernel(
    const float* __restrict__ conv_w, v2f* __restrict__ apack)
{
    const int idx = blockIdx.x * 256 + threadIdx.x;
    if (idx >= 122 * 96) return;
    const int m = idx % 96;
    const int t = idx / 96;
    v2f v;
#pragma unroll
    for (int j = 0; j < 2; ++j) {
        const int k = 2 * t + j;
        float w = 0.0f;
        if (k < KDIM && m < MOUT) {
            const int c = k / 9;
            const int p = k - c * 9;
            w = conv_w[(size_t)(m * CHn + c) * 9 + p];
        }
        v[j] = w;
    }
    apack[idx] = v;
}

// ---------------------------------------------------------------------------
// Kernel 1: 3x3 conv as implicit GEMM via V_WMMA_F32_16X16X4_F32.
//   D[96 x 64pix] = W[96 x 244] * G[244 x 64pix] per workgroup.
// Guide halo tile staged in LDS (zero-padded), B built with 2 ds loads/step,
// A read as packed float2 (coalesced b64, L2-resident).
// ---------------------------------------------------------------------------
__global__ __launch_bounds__(128) void conv3x3_wmma_kernel(
    const float* __restrict__ guide,     // [B,27,H,W]
    const v2f*   __restrict__ apack,     // [122*96] packed A fragments
    const float* __restrict__ conv_b,    // [81]
    float* __restrict__ conv_out)        // [81, NPIX]
{
    __shared__ float gtile[CHn * 3 * GS];   // 5346 floats, zero-padded halo
    __shared__ int   boff[KPAD];            // k -> LDS base offset

    const int tid = threadIdx.x;
    const int bx  = blockIdx.x;
    const int tx  = bx % NBX;
    const int yy  = (bx / NBX) % Hn;
    const int bb  = bx / (NBX * Hn);
    const int x0  = tx * TW;

    // ---- stage guide halo tile (27 ch x 3 rows x 66 cols), zeros at borders
    for (int idx = tid; idx < CHn * 3 * GS; idx += 128) {
        const int c   = idx / (3 * GS);
        const int r   = (idx - c * 3 * GS) / GS;
        const int col = idx - c * 3 * GS - r * GS;
        const int gy  = yy + r - 1;
        const int gx  = x0 + col - 1;
        float v = 0.0f;
        if ((unsigned)gy < (unsigned)Hn && (unsigned)gx < (unsigned)Wn)
            v = guide[((size_t)(bb * CHn + c) * Hn + gy) * Wn + gx];
        gtile[idx] = v;
    }
    // ---- k -> LDS offset table (k=243 pad maps anywhere; A column is zero)
    for (int k = tid; k < KPAD; k += 128) {
        const int kk = (k < KDIM) ? k : 0;
        const int c  = kk / 9;
        const int p  = kk - c * 9;
        boff[k] = (c * 3 + p / 3) * GS + (p - (p / 3) * 3);
    }
    __syncthreads();

    const int lane = tid & 31;
    const int wv   = tid >> 5;          // wave 0..3
    const int nl   = lane & 15;
    const int kh   = lane >> 4;         // K-half selector
    const int xl   = wv * 16 + nl;      // pixel within tile, 0..63

    v8f acc[MTILES];
#pragma unroll
    for (int mt = 0; mt < MTILES; ++mt) acc[mt] = (v8f){};

    for (int s = 0; s < KSTEPS; ++s) {
        const int t  = s * 2 + kh;
        const int o0 = boff[2 * t];
        const int o1 = boff[2 * t + 1];
        v2f bfrag;
        bfrag[0] = gtile[o0 + xl];
        bfrag[1] = gtile[o1 + xl];
        const v2f* arow = apack + (size_t)t * 96;
#pragma unroll
        for (int mt = 0; mt < MTILES; ++mt) {
            const v2f afrag = arow[mt * 16 + nl];
            acc[mt] = __builtin_amdgcn_wmma_f32_16x16x4_f32(
                false, afrag, false, bfrag, (short)0, acc[mt], false, false);
        }
    }

    // ---- store D + bias: lane holds D[m = j + 8*kh][n-col = nl]
    const int n = bb * (Hn * Wn) + yy * Wn + x0 + xl;
#pragma unroll
    for (int mt = 0; mt < MTILES; ++mt) {
#pragma unroll
        for (int j = 0; j < 8; ++j) {
            const int m = mt * 16 + kh * 8 + j;
            if (m < MOUT)
                conv_out[(size_t)m * NPIX + n] = acc[mt][j] + conv_b[m];
        }
    }
}

// ---------------------------------------------------------------------------
// Bilinear sample with zeros padding (matches grid_sample zeros semantics)
// ---------------------------------------------------------------------------
__device__ __forceinline__ float fetch_z(const float* __restrict__ img, int xi, int yi)
{
    if ((unsigned)xi < (unsigned)Wn && (unsigned)yi < (unsigned)Hn)
        return img[yi * Wn + xi];
    return 0.0f;
}

// ---------------------------------------------------------------------------
// Kernel 2: one propagation iteration (softmax over 9 taps fused).
// ---------------------------------------------------------------------------
__global__ __launch_bounds__(256) void dyspn_iter_kernel(
    const float* __restrict__ cur_in,      // [B,H,W]
    const float* __restrict__ conv_out,    // [81, NPIX] (bias included)
    const float* __restrict__ confidence,  // [B,1,H,W]
    const float* __restrict__ sp_dep,      // [B,1,H,W]
    const float* __restrict__ tap,         // [9,2]
    const float* __restrict__ feat_src,    // input (iter 0) or null
    float* __restrict__ cur_out,
    float* __restrict__ pred_out,          // null except last iter
    float* __restrict__ feat_out,          // null except first iter
    int iter)
{
    const int n = blockIdx.x * 256 + threadIdx.x;
    if (n >= NPIX) return;

    const int b   = n / (Hn * Wn);
    const int rem = n - b * (Hn * Wn);
    const int y   = rem / Wn;
    const int x   = rem - y * Wn;

    const float spd = sp_dep[n];
    const float cv  = confidence[n];
    const float sig = 1.0f / (1.0f + expf(-cv));
    const float sgn = (spd > 0.0f) ? 1.0f : ((spd < 0.0f) ? -1.0f : 0.0f);
    const float conf = sig * sgn;

    float aff[9];
    float mx = -3.0e38f;
#pragma unroll
    for (int t = 0; t < 9; ++t) {
        aff[t] = conv_out[(size_t)(2 * CHn + iter * 9 + t) * NPIX + n];
        mx = fmaxf(mx, aff[t]);
    }
    float ssum = 0.0f;
#pragma unroll
    for (int t = 0; t < 9; ++t) { aff[t] = expf(aff[t] - mx); ssum += aff[t]; }
    const float inv = 1.0f / ssum;

    const float* img = cur_in + (size_t)b * Hn * Wn;
    float agg = 0.0f;
#pragma unroll
    for (int t = 0; t < 9; ++t) {
        const int chx = (iter * 9 + t) * 2;
        const float px = conv_out[(size_t)chx * NPIX + n]       + tap[t * 2 + 0] + (float)x;
        const float py = conv_out[(size_t)(chx + 1) * NPIX + n] + tap[t * 2 + 1] + (float)y;

        const float x0f = floorf(px), y0f = floorf(py);
        const float wx = px - x0f,    wy = py - y0f;
        const int   x0 = (int)x0f,    y0 = (int)y0f;

        const float v00 = fetch_z(img, x0,     y0);
        const float v01 = fetch_z(img, x0 + 1, y0);
        const float v10 = fetch_z(img, x0,     y0 + 1);
        const float v11 = fetch_z(img, x0 + 1, y0 + 1);

        const float v = (1.0f - wy) * ((1.0f - wx) * v00 + wx * v01)
                      +         wy  * ((1.0f - wx) * v10 + wx * v11);
        agg += v * (aff[t] * inv);
    }

    const float outv = (1.0f - conf) * agg + conf * spd;
    cur_out[n] = outv;
    if (pred_out) pred_out[n] = outv;
    if (feat_out) feat_out[n] = feat_src[n];
}

// ---------------------------------------------------------------------------
extern "C" void kernel_launch(void* const* d_in, const int* in_sizes, int n_in,
                              void* d_out, int out_size, void* d_ws, size_t ws_size,
                              hipStream_t stream)
{
    const float* input      = (const float*)d_in[0];
    const float* guide      = (const float*)d_in[1];
    const float* sp_dep     = (const float*)d_in[2];
    const float* confidence = (const float*)d_in[3];
    const float* conv_w     = (const float*)d_in[4];
    const float* conv_b     = (const float*)d_in[5];
    const float* tap        = (const float*)d_in[6];

    float* out     = (float*)d_out;
    float* convbuf = (float*)d_ws;                       // 81 * NPIX floats
    v2f*   apack   = (v2f*)(convbuf + (size_t)MOUT * NPIX); // 122*96 float2

    float* pred   = out;
    float* feat   = out + (size_t)NPIX;
    float* inter0 = out + (size_t)2 * NPIX;
    float* inter1 = out + (size_t)3 * NPIX;
    float* inter2 = out + (size_t)4 * NPIX;

    // one-shot weight repack: 11712 float2 elements
    repack_weights_kernel<<<(122 * 96 + 255) / 256, 256, 0, stream>>>(conv_w, apack);

    // conv: 2 batches * 240 rows * 19 tiles = 9120 blocks of 128 threads
    conv3x3_wmma_kernel<<<Bn * Hn * NBX, 128, 0, stream>>>(guide, apack, conv_b, convbuf);

    const int ib = NPIX / 256; // 2280, exact
    dyspn_iter_kernel<<<ib, 256, 0, stream>>>(input,  convbuf, confidence, sp_dep, tap,
                                              input,  inter0, nullptr, feat,   0);
    dyspn_iter_kernel<<<ib, 256, 0, stream>>>(inter0, convbuf, confidence, sp_dep, tap,
                                              nullptr, inter1, nullptr, nullptr, 1);
    dyspn_iter_kernel<<<ib, 256, 0, stream>>>(inter1, convbuf, confidence, sp_dep, tap,
                                              nullptr, inter2, pred,    nullptr, 2);
}